// AELoss_69750268887702
// MI455X (gfx1250) — compile-verified
//
#include <hip/hip_runtime.h>
#include <hip/hip_bf16.h>

typedef float v2f __attribute__((ext_vector_type(2)));
typedef float v8f __attribute__((ext_vector_type(8)));

#define NH    30
#define NJ    17
#define KPAD  20            // NJ padded to multiple of 4 for WMMA K-chunks
#define PAIRS (NH * NJ)     // 510
#define BLOCK 512

// One block per batch element. Computes l0(b) + l1(b).
__global__ __launch_bounds__(BLOCK) void ae_loss_kernel(
    const float* __restrict__ tmap0,   // (B, 17, 256, 256) f32
    const float* __restrict__ tmap1,   // (B, 17, 512, 512) f32
    const int*   __restrict__ kps0,    // (B, 30, 51) i32
    const int*   __restrict__ kps1,    // (B, 30, 51) i32
    float*       __restrict__ out)     // (B,) f32
{
    const int b = blockIdx.x;
    const int t = threadIdx.x;

    __shared__ float    s_w[32][KPAD];   // masked gathered vals, zero padded (rows 30,31 = 0)
    __shared__ unsigned s_vis[NH];       // per-human visibility bitmask (17 bits)
    __shared__ float    s_sum[32];       // per-human masked row sums (from WMMA)
    __shared__ float    s_avg[32];       // per-human means (0 if empty)
    __shared__ float    s_red[BLOCK];    // deterministic tree-reduction buffer

    float total    = 0.0f;   // thread 0 accumulates final loss
    float pull_acc = 0.0f;   // thread 0 scratch
    float nh       = 0.0f;   // thread 0 scratch

    for (int lvl = 0; lvl < 2; ++lvl) {
        const float* tmap = lvl ? tmap1 : tmap0;
        const int*   kps  = lvl ? kps1  : kps0;
        const int    R    = lvl ? 512   : 256;

        // ---- zero staging LDS ----
        for (int i = t; i < 32 * KPAD; i += BLOCK) ((float*)s_w)[i] = 0.0f;
        if (t < NH) s_vis[t] = 0u;
        __syncthreads();

        // ---- gather: one thread per (human, joint) pair ----
        if (t < PAIRS) {
            const int h = t / NJ, j = t - h * NJ;
            const int* kp = kps + ((size_t)b * NH + h) * (3 * NJ) + 3 * j;
            const int x = kp[0], y = kp[1], vis = kp[2];
            float val = 0.0f;
            if (vis != 0) {
                val = tmap[(((size_t)b * NJ + j) * R + x) * R + y];
                atomicOr(&s_vis[h], 1u << j);   // bitwise OR: deterministic
            }
            s_w[h][j] = val;                    // masked value (0 if not visible)
        }
        __syncthreads();

        // ---- per-human masked sums via V_WMMA_F32_16X16X4_F32 on wave 0 ----
        // D = A(16x4) * ones(4x16); D[m,n] = sum_k A[m,k].  Chained C across 5 K-chunks.
        // A layout (ISA 7.12.2): lanes 0-15 hold K=0(a.x),K=1(a.y); lanes 16-31 hold K=2,K=3.
        // D layout: lanes 0-15, vgpr r -> row r; lanes 16-31, vgpr r -> row r+8.
        if (t < 32) {                      // wave 0: EXEC all ones as WMMA requires
            const int lane = t;
            const int rrow = lane & 15;
            const int rsel = (lane < 16) ? 0 : 2;
            v2f ones; ones.x = 1.0f; ones.y = 1.0f;
            #pragma unroll
            for (int blk = 0; blk < 2; ++blk) {
                const int m0 = blk * 16;   // rows [m0, m0+16)
                v8f c = {};
                #pragma unroll
                for (int k0 = 0; k0 < KPAD; k0 += 4) {
                    v2f a;
                    a.x = s_w[m0 + rrow][k0 + rsel];
                    a.y = s_w[m0 + rrow][k0 + rsel + 1];
                    c = __builtin_amdgcn_wmma_f32_16x16x4_f32(
                            false, a, false, ones, (short)0, c, false, false);
                }
                if (lane == 0) {
                    #pragma unroll
                    for (int i = 0; i < 8; ++i) s_sum[m0 + i] = c[i];
                } else if (lane == 16) {
                    #pragma unroll
                    for (int i = 0; i < 8; ++i) s_sum[m0 + 8 + i] = c[i];
                }
            }
        }
        __syncthreads();

        // ---- per-human mean; #valid humans (thread 0) ----
        if (t < NH) {
            const int cnt = __popc(s_vis[t]);
            s_avg[t] = (cnt > 0) ? (s_sum[t] / (float)cnt) : 0.0f;
        }
        if (t == 0) {
            nh = 0.0f;
            for (int h = 0; h < NH; ++h) nh += (s_vis[h] != 0u) ? 1.0f : 0.0f;
        }
        __syncthreads();

        // ---- pull term: sum over (h,j) of mask * (val - avg[h])^2 ----
        float p = 0.0f;
        if (t < PAIRS) {
            const int h = t / NJ, j = t - h * NJ;
            if ((s_vis[h] >> j) & 1u) {
                const float d = s_w[h][j] - s_avg[h];
                p = d * d;
            }
        }
        s_red[t] = p;
        __syncthreads();
        #pragma unroll 1
        for (int s = BLOCK / 2; s > 0; s >>= 1) {   // fixed-order: deterministic
            if (t < s) s_red[t] += s_red[t + s];
            __syncthreads();
        }
        if (t == 0) pull_acc = s_red[0];

        // ---- push term: sum over 30x30 pairs of exp(-(avg_i - avg_k)^2 / 2) ----
        float q;
        {
            const int i0 = t / NH, k0 = t - i0 * NH;           // pair t (t < 512 < 900)
            const float d0 = s_avg[i0] - s_avg[k0];
            q = __expf(-0.5f * d0 * d0);
            const int p1 = t + BLOCK;                          // pair t+512
            if (p1 < NH * NH) {
                const int i1 = p1 / NH, k1 = p1 - i1 * NH;
                const float d1 = s_avg[i1] - s_avg[k1];
                q += __expf(-0.5f * d1 * d1);
            }
        }
        s_red[t] = q;
        __syncthreads();
        #pragma unroll 1
        for (int s = BLOCK / 2; s > 0; s >>= 1) {
            if (t < s) s_red[t] += s_red[t + s];
            __syncthreads();
        }
        if (t == 0) total += pull_acc / nh + s_red[0] / (nh * nh);
        __syncthreads();   // protect s_red / s_avg / s_w before next level reuses them
    }

    if (t == 0) out[b] = total;
}

extern "C" void kernel_launch(void* const* d_in, const int* in_sizes, int n_in,
                              void* d_out, int out_size, void* d_ws, size_t ws_size,
                              hipStream_t stream) {
    const float* tmap0 = (const float*)d_in[0];
    const float* tmap1 = (const float*)d_in[1];
    const int*   kps0  = (const int*)d_in[2];
    const int*   kps1  = (const int*)d_in[3];
    float*       out   = (float*)d_out;

    const int B = in_sizes[2] / (NH * 3 * NJ);   // = 8 for the reference shapes

    ae_loss_kernel<<<B, BLOCK, 0, stream>>>(tmap0, tmap1, kps0, kps1, out);
}